// S4D_49898930045196
// MI455X (gfx1250) — compile-verified
//
#include <hip/hip_runtime.h>

// S4D forward on gfx1250 via chunked state-space (SSD) decomposition.
// One workgroup per channel d (1024 WGs), one wave32 per batch b (4 waves).
// fp32 V_WMMA_F32_16X16X4_F32 (K=16 via 4 chained WMMAs) for precision parity
// with the fp32 reference. Problem is HBM-bound (~128MB -> ~5.5us floor at
// 23.3 TB/s), so all bulk data movement goes through the Tensor Data Mover:
//   tensor_load_to_lds  : x tile (1 x B=4 x L=256), double-buffered
//   tensor_store_from_lds: silu output tile, same shape
// Steady state = WMMA + DS + TDM only; zero per-lane global vmem in the loop.

typedef float v2f __attribute__((ext_vector_type(2)));
typedef float v8f __attribute__((ext_vector_type(8)));
typedef unsigned int u32x4 __attribute__((ext_vector_type(4)));
typedef int i32x4 __attribute__((ext_vector_type(4)));
typedef int i32x8 __attribute__((ext_vector_type(8)));

#define NSTATE 16
#define CHUNK  16          // timesteps per chunk (= WMMA M)
#define NCH    16          // chunks per superstep (= WMMA N)
#define SUPER  (CHUNK*NCH) // 256 timesteps per superstep
#define SEQLEN 4096
#define BSZ    4
#define DMODEL 1024
#define NSUPER (SEQLEN / SUPER) // 16
#define S4D_SCALE 0.25f         // sqrt(1/16)

static __device__ __forceinline__ v8f wmma4(v2f a, v2f b, v8f c) {
  // D = A(16x4,f32) * B(4x16,f32) + C(16x16,f32)
  return __builtin_amdgcn_wmma_f32_16x16x4_f32(false, a, false, b,
                                               (short)0, c, false, false);
}

// Issue a TDM transfer of a (1 x BSZ x SUPER) f32 tile between LDS (dense
// [l][b]) and global x/out at element offset (l*BSZ*DMODEL + b*DMODEL + d):
// dim0: 1 contiguous elem; dim1: b, stride DMODEL; dim2: l, stride BSZ*DMODEL.
// D# encoding per CDNA5 ISA sec 8.3-8.6.
static __device__ __forceinline__ void tdm_tile(bool is_store,
                                                unsigned lds_addr,
                                                const void* gptr) {
  const unsigned long long ga = (unsigned long long)gptr;
  u32x4 g0;
  g0[0] = 1u;                                            // count=1, user mode
  g0[1] = lds_addr;                                      // lds_addr[63:32]
  g0[2] = (unsigned)(ga & 0xFFFFFFFFull);                // global_addr lo
  g0[3] = (unsigned)((ga >> 32) & 0x01FFFFFFull)         // global_addr[56:32]
          | (2u << 30);                                  // type=2 (image)
  i32x8 g1;
  g1[0] = (2 << 16);                       // wg_mask=0, data_size=2 (4 bytes)
  g1[1] = (1 << 16);                       // tensor_dim0 = 1   (bits 79:48 lo)
  g1[2] = (BSZ << 16);                     // tensor_dim1 = 4   (bits 111:80 lo)
  g1[3] = (1 << 16);                       // tile_dim0 = 1     (bits 127:112)
  g1[4] = BSZ | (SUPER << 16);             // tile_dim1=4, tile_dim2=256
  g1[5] = DMODEL;                          // tensor_dim0_stride = 1024
  g1[6] = (int)(((unsigned)(BSZ * DMODEL)) << 16);  // tensor_dim1_stride=4096
  g1[7] = 0;
  i32x4 g2;
  g2[0] = SUPER;                           // tensor_dim2 = 256
  g2[1] = 0; g2[2] = 0; g2[3] = 0;         // no dim3
  i32x4 g3; g3[0] = 0; g3[1] = 0; g3[2] = 0; g3[3] = 0;
#if __clang_major__ >= 23
  i32x8 gz; for (int i = 0; i < 8; ++i) gz[i] = 0;
  if (is_store) __builtin_amdgcn_tensor_store_from_lds(g0, g1, g2, g3, gz, 0);
  else          __builtin_amdgcn_tensor_load_to_lds (g0, g1, g2, g3, gz, 0);
#else
  if (is_store) __builtin_amdgcn_tensor_store_from_lds(g0, g1, g2, g3, 0);
  else          __builtin_amdgcn_tensor_load_to_lds (g0, g1, g2, g3, 0);
#endif
}

__launch_bounds__(128)
__global__ void s4d_fused_kernel(const float* __restrict__ x,
                                 const float* __restrict__ log_dt,
                                 const float* __restrict__ log_A_real,
                                 const float* __restrict__ A_imag,
                                 const float* __restrict__ Bp,
                                 const float* __restrict__ Cp,
                                 const float* __restrict__ Dp,
                                 float* __restrict__ out) {
  // per-channel 16x16 operator matrices (A-matrix operands of the WMMAs)
  __shared__ float Mlds[16][16];   // Toeplitz of k[0..15]            (real)
  __shared__ float Urs [16][16];   // Re(p*q^{l+1})
  __shared__ float nUis[16][16];   // -Im(p*q^{l+1}) (pre-negated: f32 WMMA has no A-neg)
  __shared__ float Wrs [16][16];   // Re(q^{15-m})
  __shared__ float Wis [16][16];   // Im(q^{15-m})
  __shared__ float q16r[16], q16i[16], prs[16], kv[16];
  // per-wave scratch for the chunk-carry scan
  __shared__ float Gr[BSZ][16][16], Gi[BSZ][16][16];
  __shared__ float Sr[BSZ][16][16], Si[BSZ][16][16];
  // TDM double buffers: x tile in, silu(y) tile out, dense [l][b]
  __shared__ float xt[2][SUPER][BSZ];
  __shared__ float yt[2][SUPER][BSZ];

  const int d   = blockIdx.x;
  const int tid = threadIdx.x;

  // ---------------- phase 0: build per-channel matrices ----------------
  if (tid < NSTATE) {
    const int n  = tid;
    const float dt  = __expf(log_dt[d]);
    const float Are = -__expf(log_A_real[d * NSTATE + n]);
    const float Aim = A_imag[d * NSTATE + n];
    const float dtAr = Are * dt, dtAi = Aim * dt;
    const float denr = 1.f - 0.5f * dtAr, deni = -0.5f * dtAi;  // 1 - dtA/2
    const float numr = 1.f + 0.5f * dtAr, numi =  0.5f * dtAi;  // 1 + dtA/2
    const float inv  = 1.f / (denr * denr + deni * deni);
    const float qr = (numr * denr + numi * deni) * inv;
    const float qi = (numi * denr - numr * deni) * inv;
    const float br = Bp[(d * NSTATE + n) * 2 + 0], bi = Bp[(d * NSTATE + n) * 2 + 1];
    const float cr = Cp[(d * NSTATE + n) * 2 + 0], ci = Cp[(d * NSTATE + n) * 2 + 1];
    const float ccr = br * cr - bi * ci, cci = br * ci + bi * cr;
    const float tr = ccr * dt, ti = cci * dt;                    // Cc*dt
    const float pr = (tr * denr + ti * deni) * inv * S4D_SCALE;  // p = Cc*dt/den*SCALE
    const float pi = (ti * denr - tr * deni) * inv * S4D_SCALE;
    prs[n] = pr;
    float er = 1.f, ei = 0.f;                                    // q^e, e=0
    for (int e = 0; e <= 16; ++e) {
      if (e >= 1) {                       // U[e-1][n] = p * q^e
        Urs [e - 1][n] =  pr * er - pi * ei;
        nUis[e - 1][n] = -(pr * ei + pi * er);
      }
      if (e <= 15) { Wrs[n][15 - e] = er; Wis[n][15 - e] = ei; } // W[n][m]=q^{15-m}
      if (e == 16) { q16r[n] = er; q16i[n] = ei; }
      const float nr = er * qr - ei * qi, ni = er * qi + ei * qr;
      er = nr; ei = ni;
    }
  }
  // kick off the first TDM x-tile load while matrices are being built
  const int wave = tid >> 5;        // batch handled by this wave
  if (wave == 0)
    tdm_tile(false, (unsigned)(unsigned long long)&xt[0][0][0], &x[d]);

  __syncthreads();
  if (tid < 16) {        // k[0]=Re(sum p); k[t]=sum_n Re(p q^t)=sum_n Ur[t-1][n]
    float acc = 0.f;
    if (tid == 0) { for (int n = 0; n < 16; ++n) acc += prs[n]; }
    else          { for (int n = 0; n < 16; ++n) acc += Urs[tid - 1][n]; }
    kv[tid] = acc;
  }
  __syncthreads();
  for (int i = tid; i < 256; i += blockDim.x) {
    const int l = i >> 4, m = i & 15;
    Mlds[l][m] = (l >= m) ? kv[l - m] : 0.f;
  }
  __syncthreads();

  // ---------------- phase 1: per-wave (d,b) sweep over L ----------------
  const int lane = tid & 31;
  const int h    = lane >> 4;       // wave half (K-pair select)
  const int j    = lane & 15;       // A row / B column / C column
  const int kb   = 2 * h;
  const int b    = wave;
  const float Dpd = Dp[d];

  // A-matrix operands (ISA 7.12.2: lanes 0-15 rows M, VGPR0/1 = K pair {2h,2h+1})
  v2f aM[4], aUr[4], aUi[4], aWr[4], aWi[4];
  for (int k = 0; k < 4; ++k) {
    const int c0 = 4 * k + kb;
    aM [k].x = Mlds[j][c0]; aM [k].y = Mlds[j][c0 + 1];
    aUr[k].x = Urs [j][c0]; aUr[k].y = Urs [j][c0 + 1];
    aUi[k].x = nUis[j][c0]; aUi[k].y = nUis[j][c0 + 1];
    aWr[k].x = Wrs [j][c0]; aWr[k].y = Wrs [j][c0 + 1];
    aWi[k].x = Wis [j][c0]; aWi[k].y = Wis [j][c0 + 1];
  }
  const float qbr = q16r[j], qbi = q16i[j];  // q^16 for state n=j (lanes 0-15)

  float scr = 0.f, sci = 0.f;                // carried complex state (lanes 0-15)

  for (int blk = 0; blk < NSUPER; ++blk) {
    const int buf = blk & 1;

    // TDM pipeline (wave 0): issue load(blk+1); guarantee load(blk) and
    // store(blk-2) complete. Issue order is L0,L1,S0,L2,S1,... so allowing
    // 2 outstanding (L(blk+1), S(blk-1)) is exact; 1 at the pipe edges.
    if (wave == 0) {
      if (blk + 1 < NSUPER)
        tdm_tile(false, (unsigned)(unsigned long long)&xt[buf ^ 1][0][0],
                 &x[(size_t)(blk + 1) * SUPER * (BSZ * DMODEL) + d]);
      if (blk == 0 || blk + 1 == NSUPER)
        __builtin_amdgcn_s_wait_tensorcnt(1);
      else
        __builtin_amdgcn_s_wait_tensorcnt(2);
    }
    __syncthreads();   // xt[buf] ready, yt[buf] free, for all waves

    // X in B-matrix layout: lane(h,j) step k holds X[4k+2h][j], X[4k+2h+1][j]
    // where X[t][col] = x[blk*256 + col*16 + t] for this (d,b)  (LDS reads)
    v2f xb[4];
    for (int k = 0; k < 4; ++k) {
      const int t0 = 4 * k + kb;
      xb[k].x = xt[buf][j * CHUNK + t0][b];
      xb[k].y = xt[buf][j * CHUNK + t0 + 1][b];
    }

    // G = W @ X   (complex 16x16 = two real matmuls, X real)
    v8f gr, gi;
    for (int r = 0; r < 8; ++r) { gr[r] = 0.f; gi[r] = 0.f; }
    for (int k = 0; k < 4; ++k) {
      gr = wmma4(aWr[k], xb[k], gr);
      gi = wmma4(aWi[k], xb[k], gi);
    }
    // spill G (C/D layout: vgpr r = row 8h+r, column j)
    for (int r = 0; r < 8; ++r) {
      Gr[wave][8 * h + r][j] = gr[r];
      Gi[wave][8 * h + r][j] = gi[r];
    }
    __syncthreads();

    // serial chunk-carry scan: S[:,c] = state entering chunk c
    if (lane < 16) {
      float sr = scr, si = sci;
      for (int c = 0; c < 16; ++c) {
        Sr[wave][j][c] = sr; Si[wave][j][c] = si;
        const float g_r = Gr[wave][j][c], g_i = Gi[wave][j][c];
        const float nsr = qbr * sr - qbi * si + g_r;
        const float nsi = qbr * si + qbi * sr + g_i;
        sr = nsr; si = nsi;
      }
      scr = sr; sci = si;
    }
    __syncthreads();

    // S in B-matrix layout
    v2f sbr[4], sbi[4];
    for (int k = 0; k < 4; ++k) {
      const int r0 = 4 * k + kb;
      sbr[k].x = Sr[wave][r0][j];     sbr[k].y = Sr[wave][r0 + 1][j];
      sbi[k].x = Si[wave][r0][j];     sbi[k].y = Si[wave][r0 + 1][j];
    }

    // Y = M@X + Ur@Sr + (-Ui)@Si   (accumulated in one fp32 tile)
    v8f y;
    for (int r = 0; r < 8; ++r) y[r] = 0.f;
    for (int k = 0; k < 4; ++k) y = wmma4(aM [k], xb [k], y);
    for (int k = 0; k < 4; ++k) y = wmma4(aUr[k], sbr[k], y);
    for (int k = 0; k < 4; ++k) y = wmma4(aUi[k], sbi[k], y);

    // fused epilogue: silu(y + x*D_param) -> LDS out tile (v_rcp_f32, no div)
    for (int r = 0; r < 8; ++r) {
      const int ll  = j * CHUNK + 8 * h + r;     // local timestep in superstep
      const float v = y[r] + xt[buf][ll][b] * Dpd;
      const float sig = __builtin_amdgcn_rcpf(1.f + __expf(-v));
      yt[buf][ll][b] = v * sig;
    }
    __syncthreads();   // yt[buf] complete across all waves

    if (wave == 0)
      tdm_tile(true, (unsigned)(unsigned long long)&yt[buf][0][0],
               &out[(size_t)blk * SUPER * (BSZ * DMODEL) + d]);
  }
  // final stores drain via S_ENDPGM's implicit wait-idle; make it explicit:
  if (wave == 0) __builtin_amdgcn_s_wait_tensorcnt(0);
}

extern "C" void kernel_launch(void* const* d_in, const int* in_sizes, int n_in,
                              void* d_out, int out_size, void* d_ws, size_t ws_size,
                              hipStream_t stream) {
  (void)in_sizes; (void)n_in; (void)d_ws; (void)ws_size; (void)out_size;
  const float* x          = (const float*)d_in[0];
  const float* log_dt     = (const float*)d_in[1];
  const float* log_A_real = (const float*)d_in[2];
  const float* A_imag     = (const float*)d_in[3];
  const float* Bp         = (const float*)d_in[4];
  const float* Cp         = (const float*)d_in[5];
  const float* Dp         = (const float*)d_in[6];
  float* out = (float*)d_out;

  s4d_fused_kernel<<<dim3(DMODEL), dim3(128), 0, stream>>>(
      x, log_dt, log_A_real, A_imag, Bp, Cp, Dp, out);
}